// GINENet_24464133718762
// MI455X (gfx1250) — compile-verified
//
#include <hip/hip_runtime.h>
#include <hip/hip_bf16.h>

#define NNODES 50000
#define NEDGES 800000
#define DFEAT  128
#define BN_EPS 1e-5f
#define KP     256      // stride between sum[] and sumsq[] in stats buffer
#define KINMAX 192
#define LDSPAD 4        // +4 floats/row: lanes reading col*stride hit distinct LDS banks

typedef float v2f __attribute__((ext_vector_type(2)));
typedef float v8f __attribute__((ext_vector_type(8)));

// ---------------------------------------------------------------- utilities
__global__ void copy_f32(const float* __restrict__ in, float* __restrict__ out, int n) {
    int i = blockIdx.x * blockDim.x + threadIdx.x;
    if (i < n) out[i] = in[i];
}

__global__ void zero_f32(float* __restrict__ p, int n) {
    int i = blockIdx.x * blockDim.x + threadIdx.x;
    if (i < n) p[i] = 0.f;
}

// -------------------------------------------------- GINE aggregation (edge scatter)
// out must be pre-initialized with X (self term).  msg = relu(x[src] + ea*elw + elb)
// One thread handles 4 contiguous features of one edge (D=128 -> 32 threads/edge).
__global__ void gine_aggr(const float* __restrict__ X,
                          const long long* __restrict__ src,
                          const long long* __restrict__ dst,
                          const float* __restrict__ ea,
                          const float* __restrict__ elw,
                          const float* __restrict__ elb,
                          float* __restrict__ out, int E) {
    int tid = blockIdx.x * blockDim.x + threadIdx.x;
    int total = E * (DFEAT >> 2);                 // 25.6M threads
    if (tid >= total) return;
    int j = tid >> 5;                             // edge index (DFEAT/4 == 32)
    int q = (tid & 31) << 2;                      // feature offset
    long long sN = src[j];
    long long dN = dst[j];
    float a = ea[j];
    const float4 xv = *(const float4*)(X + (size_t)sN * DFEAT + q);
    const float4 wv = *(const float4*)(elw + q);
    const float4 bv = *(const float4*)(elb + q);
    float4 m;
    m.x = fmaxf(xv.x + fmaf(a, wv.x, bv.x), 0.f);
    m.y = fmaxf(xv.y + fmaf(a, wv.y, bv.y), 0.f);
    m.z = fmaxf(xv.z + fmaf(a, wv.z, bv.z), 0.f);
    m.w = fmaxf(xv.w + fmaf(a, wv.w, bv.w), 0.f);
    float* po = out + (size_t)dN * DFEAT + q;
    atomicAdd(po + 0, m.x);
    atomicAdd(po + 1, m.y);
    atomicAdd(po + 2, m.z);
    atomicAdd(po + 3, m.w);
}

// -------------------------------------------------- fused GEMM + bias + ReLU + BN-stat partials
// Y[M,Kout] = relu(H[M,Kin] @ W[Kout,Kin]^T + b); per-column sum / sum^2 atomically
// accumulated into stats[] for the BatchNorm pass.
// Block = 256 threads = 8 waves. Block owns one 16-row m-strip: the 16 x Kin A-tile
// is staged once in LDS (row stride Kin+4 -> conflict-free ds_load_b64), then wave w
// computes o-tile w with V_WMMA_F32_16X16X4_F32 reading A from LDS, B from global.
// GUARD=true only for the last layer (Kout=8, partial o-tile).
// ISA layouts (wave32):
//   A 16x4 : lane holds row M=lane&15; VGPR0/1 = K = (lane>=16 ? 2,3 : 0,1)
//   B 4x16 : lane holds col N=lane&15; VGPR0/1 = K = (lane>=16 ? 2,3 : 0,1)
//   C/D    : VGPR v holds M = v + (lane>=16 ? 8 : 0), N = lane&15
template <bool GUARD>
__global__ __launch_bounds__(256) void gemm_relu_stats(
        const float* __restrict__ H, const float* __restrict__ W,
        const float* __restrict__ bias, float* __restrict__ Y,
        float* __restrict__ stats, int Kin, int Kout, int numOTiles) {
    __shared__ float As[16 * (KINMAX + LDSPAD)];
    const int ldsK = Kin + LDSPAD;
    const int mBase = blockIdx.x << 4;

    // cooperative stage of A tile [16 x Kin] into LDS (float4 chunks; Kin % 4 == 0)
    const int chunks = Kin >> 2;                  // float4 chunks per row
    for (int idx = threadIdx.x; idx < 16 * chunks; idx += blockDim.x) {
        int r = idx / chunks;
        int c = (idx - r * chunks) << 2;
        *(float4*)(&As[r * ldsK + c]) = *(const float4*)(H + (size_t)(mBase + r) * Kin + c);
    }
    __syncthreads();

    int wave = threadIdx.x >> 5;
    if (wave >= numOTiles) return;                // wave-uniform: EXEC stays all-1s
    int lane = threadIdx.x & 31;

    int col = lane & 15;
    int oBase = wave << 4;
    int o = oBase + col;
    bool oValid = GUARD ? (o < Kout) : true;
    int oc = oValid ? o : 0;
    int kHalf = (lane >> 4) << 1;                 // 0 or 2

    const float* aLds = &As[col * ldsK + kHalf];
    const float* bPtr = W + (size_t)oc * Kin + kHalf;
    __builtin_prefetch(bPtr, 0, 0);

    v8f acc = {};
    for (int k = 0; k < Kin; k += 4) {
        v2f a = *(const v2f*)(aLds + k);          // ds_load_b64, conflict-free
        v2f b = *(const v2f*)(bPtr + k);          // global_load_b64
        if (GUARD) {
            b.x = oValid ? b.x : 0.f;             // value-select, no divergence
            b.y = oValid ? b.y : 0.f;
        }
        acc = __builtin_amdgcn_wmma_f32_16x16x4_f32(
            /*neg_a=*/false, a, /*neg_b=*/false, b,
            /*c_mod=*/(short)0, acc, /*reuse_a=*/false, /*reuse_b=*/false);
    }

    float bv = oValid ? bias[oc] : 0.f;
    int mOff = mBase + ((lane >> 4) << 3);        // +8 rows for upper half-wave
    float s = 0.f, s2 = 0.f;
#pragma unroll
    for (int v = 0; v < 8; ++v) {
        float val = fmaxf(acc[v] + bv, 0.f);
        if (oValid) Y[(size_t)(mOff + v) * Kout + o] = val;
        s  += val;
        s2 += val * val;
    }
    // lanes L and L+16 cover the same column (rows 0-7 / 8-15): combine, then atomics
    s  += __shfl_xor(s, 16);
    s2 += __shfl_xor(s2, 16);
    if (lane < 16 && oValid) {
        atomicAdd(&stats[o], s);
        atomicAdd(&stats[KP + o], s2);
    }
}

// -------------------------------------------------- BatchNorm normalize (in place)
__global__ void bn_normalize(float* __restrict__ Y, const float* __restrict__ stats,
                             const float* __restrict__ g, const float* __restrict__ beta,
                             int total, int Kout, float invM) {
    int i = blockIdx.x * blockDim.x + threadIdx.x;
    if (i >= total) return;
    int o = i % Kout;
    float mu  = stats[o] * invM;
    float var = fmaf(-mu, mu, stats[KP + o] * invM);   // E[h^2] - mu^2 (biased)
    float r = rsqrtf(var + BN_EPS);
    Y[i] = fmaf(g[o] * r, Y[i] - mu, beta[o]);
}

// -------------------------------------------------- concat [x1 | x2] along features
__global__ void concat_kernel(const float* __restrict__ A, const float* __restrict__ B,
                              float* __restrict__ C, int n, int da, int db) {
    int i = blockIdx.x * blockDim.x + threadIdx.x;
    int dc = da + db;
    int total = n * dc;
    if (i >= total) return;
    int r = i / dc, c = i - r * dc;
    C[i] = (c < da) ? A[(size_t)r * da + c] : B[(size_t)r * db + (c - da)];
}

// ==================================================================== launch
extern "C" void kernel_launch(void* const* d_in, const int* in_sizes, int n_in,
                              void* d_out, int out_size, void* d_ws, size_t ws_size,
                              hipStream_t stream) {
    const float*     x     = (const float*)d_in[0];
    const long long* eidx  = (const long long*)d_in[1];   // int64 [2,E]
    const float*     eattr = (const float*)d_in[2];
    const float*     elw   = (const float*)d_in[3];
    const float*     elb   = (const float*)d_in[4];
    const float *c1w = (const float*)d_in[5],  *c1b = (const float*)d_in[6],
                *c1g = (const float*)d_in[7],  *c1be = (const float*)d_in[8];
    const float *c2w = (const float*)d_in[9],  *c2b = (const float*)d_in[10],
                *c2g = (const float*)d_in[11], *c2be = (const float*)d_in[12];
    const float *l1w = (const float*)d_in[13], *l1b = (const float*)d_in[14],
                *l1g = (const float*)d_in[15], *l1be = (const float*)d_in[16];
    const float *m1w = (const float*)d_in[17], *m1b = (const float*)d_in[18],
                *m1g = (const float*)d_in[19], *m1be = (const float*)d_in[20];
    const float *m2w = (const float*)d_in[21], *m2b = (const float*)d_in[22],
                *m2g = (const float*)d_in[23], *m2be = (const float*)d_in[24];

    const long long* src = eidx;
    const long long* dst = eidx + NEDGES;

    float* ws    = (float*)d_ws;
    float* bufA  = ws;                                   // N*192
    float* bufB  = bufA + (size_t)NNODES * 192;          // N*128
    float* bufC  = bufB + (size_t)NNODES * 128;          // N*64
    float* stats = bufC + (size_t)NNODES * 64;           // 2*KP floats

    const int thr = 256;
    const int nXD = NNODES * DFEAT;
    const int nMT = NNODES / 16;                         // 3125 exact (one block per m-strip)
    const float invN = 1.f / (float)NNODES;
    const int aggrTot = NEDGES * (DFEAT / 4);

    // ---- aggr1: bufA = x + sum relu(x[src] + e)
    copy_f32<<<(nXD + thr - 1) / thr, thr, 0, stream>>>(x, bufA, nXD);
    gine_aggr<<<(aggrTot + thr - 1) / thr, thr, 0, stream>>>(x, src, dst, eattr, elw, elb, bufA, NEDGES);

    // ---- conv1 MLP: x1 = BN(relu(bufA @ c1^T + b))   [N,128] -> bufB
    {
        int Kin = 128, Kout = 128, nOT = 8, tot = NNODES * Kout;
        zero_f32<<<2, thr, 0, stream>>>(stats, 2 * KP);
        gemm_relu_stats<false><<<nMT, thr, 0, stream>>>(bufA, c1w, c1b, bufB, stats, Kin, Kout, nOT);
        bn_normalize<<<(tot + thr - 1) / thr, thr, 0, stream>>>(bufB, stats, c1g, c1be, tot, Kout, invN);
    }

    // ---- aggr2: bufA = x1 + sum relu(x1[src] + e)
    copy_f32<<<(nXD + thr - 1) / thr, thr, 0, stream>>>(bufB, bufA, nXD);
    gine_aggr<<<(aggrTot + thr - 1) / thr, thr, 0, stream>>>(bufB, src, dst, eattr, elw, elb, bufA, NEDGES);

    // ---- conv2 MLP: x2 = BN(relu(bufA @ c2^T + b))   [N,64] -> bufC
    {
        int Kin = 128, Kout = 64, nOT = 4, tot = NNODES * Kout;
        zero_f32<<<2, thr, 0, stream>>>(stats, 2 * KP);
        gemm_relu_stats<false><<<nMT, thr, 0, stream>>>(bufA, c2w, c2b, bufC, stats, Kin, Kout, nOT);
        bn_normalize<<<(tot + thr - 1) / thr, thr, 0, stream>>>(bufC, stats, c2g, c2be, tot, Kout, invN);
    }

    // ---- concat [x1|x2] -> bufA [N,192]
    {
        int tot = NNODES * 192;
        concat_kernel<<<(tot + thr - 1) / thr, thr, 0, stream>>>(bufB, bufC, bufA, NNODES, 128, 64);
    }

    // ---- lin1: h3 = BN(relu(bufA @ l1^T + b))   [N,96] -> bufB
    {
        int Kin = 192, Kout = 96, nOT = 6, tot = NNODES * Kout;
        zero_f32<<<2, thr, 0, stream>>>(stats, 2 * KP);
        gemm_relu_stats<false><<<nMT, thr, 0, stream>>>(bufA, l1w, l1b, bufB, stats, Kin, Kout, nOT);
        bn_normalize<<<(tot + thr - 1) / thr, thr, 0, stream>>>(bufB, stats, l1g, l1be, tot, Kout, invN);
    }

    // ---- mlp1 layer1: h4 = BN(relu(bufB @ m1^T + b))   [N,96] -> bufA
    {
        int Kin = 96, Kout = 96, nOT = 6, tot = NNODES * Kout;
        zero_f32<<<2, thr, 0, stream>>>(stats, 2 * KP);
        gemm_relu_stats<false><<<nMT, thr, 0, stream>>>(bufB, m1w, m1b, bufA, stats, Kin, Kout, nOT);
        bn_normalize<<<(tot + thr - 1) / thr, thr, 0, stream>>>(bufA, stats, m1g, m1be, tot, Kout, invN);
    }

    // ---- mlp1 layer2: out = BN(relu(bufA @ m2^T + b))   [N,8] -> d_out
    {
        int Kin = 96, Kout = 8, nOT = 1, tot = NNODES * Kout;
        zero_f32<<<2, thr, 0, stream>>>(stats, 2 * KP);
        gemm_relu_stats<true><<<nMT, thr, 0, stream>>>(bufA, m2w, m2b, (float*)d_out, stats, Kin, Kout, nOT);
        bn_normalize<<<(tot + thr - 1) / thr, thr, 0, stream>>>((float*)d_out, stats, m2g, m2be, tot, Kout, invN);
    }
}